// IPAGNN_88012469830544
// MI455X (gfx1250) — compile-verified
//
#include <hip/hip_runtime.h>
#include <cstddef>

typedef float v2f __attribute__((ext_vector_type(2)));
typedef float v8f __attribute__((ext_vector_type(8)));

namespace {
constexpr int kB    = 8;
constexpr int kN    = 256;
constexpr int kL    = 4;
constexpr int kH    = 128;
constexpr int kBN   = kB * kN;        // 2048
constexpr int kG    = 4 * kH;         // 512 gate width
constexpr int kSteps = 16;
constexpr int kOutV = 30000;
constexpr size_t kHOFF = (size_t)2 * kBN * kH;  // h-region offset inside a state buffer
constexpr int AS = 132;               // padded LDS stride for A-operand tiles (bank-conflict free)
constexpr int GS = 516;               // padded LDS stride for gate buffer
constexpr int kPairs = 2 * (kH / 2) * kG;       // packed weight pairs per matrix: 2 layers * 64 * 512
}

__device__ __forceinline__ float sigmf(float x) { return 1.0f / (1.0f + expf(-x)); }

__device__ __forceinline__ v8f wmma4(v2f a, v2f b, v8f c) {
  // D = A(16x4,f32) * B(4x16,f32) + C(16x16,f32)
  return __builtin_amdgcn_wmma_f32_16x16x4_f32(false, a, false, b, (short)0, c, false, false);
}

// acc += A(16x128, LDS stride AS) @ W(128x512, packed as float2 of row-pairs), cols [c_base, c_base+64)
__device__ __forceinline__ void gemm_tile(const float* __restrict__ A,
                                          const v2f* __restrict__ Wp,
                                          int c_base, int lane, v8f acc[4]) {
  const int m  = lane & 15;
  const int hf = lane >> 4;
  const int n  = lane & 15;
#pragma unroll 2
  for (int kc = 0; kc < 32; ++kc) {
    const int kk = kc * 4 + 2 * hf;                 // A layout: lanes0-15 K={k0,k0+1}, lanes16-31 K={k0+2,k0+3}
    v2f a = *(const v2f*)(A + m * AS + kk);
    const v2f* Wr = Wp + (size_t)(kc * 2 + hf) * kG; // B layout: vgpr = k&1, lane-half = k>>1 -> one b64 per tile
#pragma unroll
    for (int t = 0; t < 4; ++t) {
      acc[t] = wmma4(a, Wr[c_base + t * 16 + n], acc[t]);
    }
  }
}

__global__ __launch_bounds__(256) void k_lstm_step(
    const float* __restrict__ emb, const float* __restrict__ Sin,
    float* __restrict__ Stg, const float* __restrict__ ipin,
    float* __restrict__ ptb, float* __restrict__ pfb,
    const v2f* __restrict__ Wxp, const v2f* __restrict__ Whp,
    const float* __restrict__ bias, const float* __restrict__ bW,
    const float* __restrict__ bb, const int* __restrict__ exitI) {
  __shared__ float sX[16 * AS];        // token embedding tile (A-operand layout)
  __shared__ float sH[2][16 * AS];     // h state per layer (A-operand layout)
  __shared__ float sC[2][16 * kH];     // c state per layer
  __shared__ float sG[16 * GS];        // gate matrix 16x512
  __shared__ float sLg[32];

  const int tid  = threadIdx.x;
  const int lane = tid & 31;
  const int wave = tid >> 5;           // 8 waves
  const int r0   = blockIdx.x * 16;    // 16 nodes per workgroup
  const int c_base = wave * 64;        // each wave: 64 gate columns (4 tiles)

  // load step-input states into LDS
  for (int idx = tid; idx < 2 * 16 * kH; idx += 256) {
    int l = idx >> 11, rem = idx & 2047, m = rem >> 7, k = rem & 127;
    size_t row = r0 + m;
    sC[l][m * kH + k] = Sin[((size_t)l * kBN + row) * kH + k];
    sH[l][m * AS + k] = Sin[kHOFF + ((size_t)l * kBN + row) * kH + k];
  }
  __syncthreads();

  for (int tok = 0; tok < kL; ++tok) {
    for (int idx = tid; idx < 16 * kH; idx += 256) {
      int m = idx >> 7, k = idx & 127;
      sX[m * AS + k] = emb[(((size_t)(r0 + m)) * kL + tok) * kH + k];
    }
    __syncthreads();
    for (int l = 0; l < 2; ++l) {
      v8f acc[4];
#pragma unroll
      for (int t = 0; t < 4; ++t) {
        float blv = bias[l * kG + c_base + t * 16 + (lane & 15)];
#pragma unroll
        for (int v = 0; v < 8; ++v) acc[t][v] = blv;
      }
      const float* Ain = (l == 0) ? sX : sH[0];
      gemm_tile(Ain,   Wxp + (size_t)l * (kH / 2) * kG, c_base, lane, acc);
      gemm_tile(sH[l], Whp + (size_t)l * (kH / 2) * kG, c_base, lane, acc);
      // scatter gates to LDS (C-matrix layout -> row-major)
      const int hf = lane >> 4;
#pragma unroll
      for (int t = 0; t < 4; ++t) {
        int col = c_base + t * 16 + (lane & 15);
#pragma unroll
        for (int v = 0; v < 8; ++v) sG[(v + 8 * hf) * GS + col] = acc[t][v];
      }
      __syncthreads();
      // elementwise LSTM cell update
      for (int idx = tid; idx < 16 * kH; idx += 256) {
        int m = idx >> 7, k = idx & 127;
        float iv = sG[m * GS + k];
        float fv = sG[m * GS + kH + k];
        float gv = sG[m * GS + 2 * kH + k];
        float ov = sG[m * GS + 3 * kH + k];
        float cold = sC[l][m * kH + k];
        float cn = sigmf(fv) * cold + sigmf(iv) * tanhf(gv);
        float hn = sigmf(ov) * tanhf(cn);
        sC[l][m * kH + k] = cn;
        sH[l][m * AS + k] = hn;
      }
      __syncthreads();
    }
  }

  // exit-node preservation: restore step-input state at n == exit_index[b]
  const int bidx = r0 >> 8;
  const int mm = exitI[bidx] - (r0 & 255);
  if (mm >= 0 && mm < 16) {
    for (int idx = tid; idx < 2 * kH; idx += 256) {
      int l = idx >> 7, k = idx & 127;
      size_t row = r0 + mm;
      sC[l][mm * kH + k] = Sin[((size_t)l * kBN + row) * kH + k];
      sH[l][mm * AS + k] = Sin[kHOFF + ((size_t)l * kBN + row) * kH + k];
    }
  }
  __syncthreads();

  // branch logits: concat = [c0, h0, c1, h1] (512) @ branch_W (512x2)
  if (tid < 32) {
    int m = tid >> 1, j = tid & 1;
    float s = bb[j];
    for (int k = 0; k < kH; ++k) s += sC[0][m * kH + k] * bW[k * 2 + j];
    for (int k = 0; k < kH; ++k) s += sH[0][m * AS + k] * bW[(kH + k) * 2 + j];
    for (int k = 0; k < kH; ++k) s += sC[1][m * kH + k] * bW[(2 * kH + k) * 2 + j];
    for (int k = 0; k < kH; ++k) s += sH[1][m * AS + k] * bW[(3 * kH + k) * 2 + j];
    sLg[tid] = s;
  }
  __syncthreads();
  if (tid < 16) {
    float l0 = sLg[2 * tid], l1 = sLg[2 * tid + 1];
    float mx = fmaxf(l0, l1);
    float e0 = expf(l0 - mx), e1 = expf(l1 - mx);
    float inv = 1.0f / (e0 + e1);
    float ipv = ipin[r0 + tid];
    ptb[r0 + tid] = e0 * inv * ipv;
    pfb[r0 + tid] = e1 * inv * ipv;
  }

  // stage computed states for the aggregation kernel
  for (int idx = tid; idx < 2 * 16 * kH; idx += 256) {
    int l = idx >> 11, rem = idx & 2047, m = rem >> 7, k = rem & 127;
    size_t row = r0 + m;
    Stg[((size_t)l * kBN + row) * kH + k]         = sC[l][m * kH + k];
    Stg[kHOFF + ((size_t)l * kBN + row) * kH + k] = sH[l][m * AS + k];
  }
}

__global__ __launch_bounds__(128) void k_aggregate(
    const float* __restrict__ Stg, const float* __restrict__ Sin,
    const float* __restrict__ ipin, const float* __restrict__ ptb,
    const float* __restrict__ pfb, const int* __restrict__ ti,
    const int* __restrict__ fi, const int* __restrict__ steps,
    float* __restrict__ Sout, float* __restrict__ ipout, int step) {
  const int tgt = blockIdx.x;           // (b, m)
  const int b = tgt >> 8;
  const int m = tgt & 255;
  const int h = threadIdx.x;            // 0..127
  float aC0 = 0.f, aH0 = 0.f, aC1 = 0.f, aH1 = 0.f, aip = 0.f;
  const int base = b * kN;
  for (int n = 0; n < kN; ++n) {
    int src = base + n;
    float w = 0.f;
    if (ti[src] == m) w += ptb[src];
    if (fi[src] == m) w += pfb[src];
    if (w != 0.f) {
      aip += w;
      aC0 += w * Stg[((size_t)src) * kH + h];
      aH0 += w * Stg[kHOFF + ((size_t)src) * kH + h];
      aC1 += w * Stg[((size_t)kBN + src) * kH + h];
      aH1 += w * Stg[kHOFF + ((size_t)kBN + src) * kH + h];
    }
  }
  float denom = aip + 1e-7f;
  bool active = step < steps[b];
  size_t row = tgt;
  Sout[row * kH + h]                         = active ? aC0 / denom : Sin[row * kH + h];
  Sout[kHOFF + row * kH + h]                 = active ? aH0 / denom : Sin[kHOFF + row * kH + h];
  Sout[((size_t)kBN + row) * kH + h]         = active ? aC1 / denom : Sin[((size_t)kBN + row) * kH + h];
  Sout[kHOFF + ((size_t)kBN + row) * kH + h] = active ? aH1 / denom : Sin[kHOFF + ((size_t)kBN + row) * kH + h];
  if (h == 0) ipout[tgt] = active ? aip : ipin[tgt];
}

__global__ __launch_bounds__(256) void k_logits(
    const float* __restrict__ S, const int* __restrict__ exitI,
    const float* __restrict__ outW, const float* __restrict__ outb,
    float* __restrict__ out) {
  __shared__ float sF[kB * 512];
  const int tid = threadIdx.x;
  for (int idx = tid; idx < kB * 512; idx += 256) {
    int q = idx >> 9, k = idx & 511, seg = k >> 7, kk = k & 127;
    size_t row = (size_t)q * kN + exitI[q];
    float v;
    if (seg == 0)      v = S[row * kH + kk];
    else if (seg == 1) v = S[kHOFF + row * kH + kk];
    else if (seg == 2) v = S[((size_t)kBN + row) * kH + kk];
    else               v = S[kHOFF + ((size_t)kBN + row) * kH + kk];
    sF[idx] = v;
  }
  __syncthreads();
  int j = blockIdx.x * 256 + tid;
  if (j >= kOutV) return;
  float acc[kB];
  float bj = outb[j];
#pragma unroll
  for (int q = 0; q < kB; ++q) acc[q] = bj;
  for (int k = 0; k < 512; ++k) {
    float w = outW[(size_t)k * kOutV + j];
#pragma unroll
    for (int q = 0; q < kB; ++q) acc[q] += sF[q * 512 + k] * w;
  }
  for (int q = 0; q < kB; ++q) out[(size_t)q * kOutV + j] = acc[q];
}

__global__ void k_embed(const int* __restrict__ data, const float* __restrict__ embed,
                        float* __restrict__ embuf, int total) {
  int idx = blockIdx.x * 256 + threadIdx.x;
  if (idx >= total) return;
  int token = idx >> 7, h = idx & 127;
  embuf[idx] = embed[(size_t)data[token] * kH + h];
}

// pack W (2 layers of 128x512 row-major) into row-pair float2 fragments:
// Wp[l][p][col] = (W[l][2p][col], W[l][2p+1][col])
__global__ void k_pack(const float* __restrict__ Wsrc, float* __restrict__ Wdst, int total) {
  int idx = blockIdx.x * 256 + threadIdx.x;   // pair index (l, p, col)
  if (idx >= total) return;
  int col = idx & 511;
  int p   = (idx >> 9) & 63;
  int l   = idx >> 15;
  const float* Wl = Wsrc + (size_t)l * kH * kG;
  v2f v;
  v.x = Wl[(size_t)(2 * p) * kG + col];
  v.y = Wl[(size_t)(2 * p + 1) * kG + col];
  *(v2f*)(Wdst + (size_t)idx * 2) = v;
}

__global__ void k_init(float* __restrict__ S0, float* __restrict__ ip0, int total) {
  int idx = blockIdx.x * 256 + threadIdx.x;
  if (idx < total) S0[idx] = 0.f;
  if (idx < kBN) ip0[idx] = ((idx & 255) == 0) ? 1.0f : 0.0f;
}

extern "C" void kernel_launch(void* const* d_in, const int* in_sizes, int n_in,
                              void* d_out, int out_size, void* d_ws, size_t ws_size,
                              hipStream_t stream) {
  const int*   data  = (const int*)d_in[0];
  const int*   ti    = (const int*)d_in[1];
  const int*   fi    = (const int*)d_in[2];
  const int*   exitI = (const int*)d_in[3];
  const int*   steps = (const int*)d_in[4];
  const float* embed = (const float*)d_in[5];
  const float* Wx    = (const float*)d_in[6];
  const float* Wh    = (const float*)d_in[7];
  const float* bias  = (const float*)d_in[8];
  const float* bW    = (const float*)d_in[9];
  const float* bb    = (const float*)d_in[10];
  const float* outW  = (const float*)d_in[11];
  const float* outb  = (const float*)d_in[12];
  float* out = (float*)d_out;

  float* ws = (float*)d_ws;
  const size_t EMB = (size_t)kBN * kL * kH;   // 1,048,576 floats
  const size_t ST  = (size_t)4 * kBN * kH;    // 1,048,576 floats (c+h, 2 layers)
  float* embuf = ws;
  float* S0  = embuf + EMB;
  float* S1  = S0 + ST;
  float* STG = S1 + ST;
  float* ip0 = STG + ST;
  float* ip1 = ip0 + kBN;
  float* ptb = ip1 + kBN;
  float* pfb = ptb + kBN;
  float* Wxp = pfb + kBN;                     // packed weights: kPairs float2 each
  float* Whp = Wxp + (size_t)kPairs * 2;

  k_embed<<<(int)((EMB + 255) / 256), 256, 0, stream>>>(data, embed, embuf, (int)EMB);
  k_init<<<(int)((ST + 255) / 256), 256, 0, stream>>>(S0, ip0, (int)ST);
  k_pack<<<(kPairs + 255) / 256, 256, 0, stream>>>(Wx, Wxp, kPairs);
  k_pack<<<(kPairs + 255) / 256, 256, 0, stream>>>(Wh, Whp, kPairs);

  float* Sin = S0; float* Sout = S1; float* ipin = ip0; float* ipout = ip1;
  for (int s = 0; s < kSteps; ++s) {
    k_lstm_step<<<kBN / 16, 256, 0, stream>>>(embuf, Sin, STG, ipin, ptb, pfb,
                                              (const v2f*)Wxp, (const v2f*)Whp,
                                              bias, bW, bb, exitI);
    k_aggregate<<<kBN, 128, 0, stream>>>(STG, Sin, ipin, ptb, pfb, ti, fi, steps,
                                         Sout, ipout, s);
    float* t1 = Sin; Sin = Sout; Sout = t1;
    float* t2 = ipin; ipin = ipout; ipout = t2;
  }
  k_logits<<<(kOutV + 255) / 256, 256, 0, stream>>>(Sin, exitI, outW, outb, out);
  (void)in_sizes; (void)n_in; (void)out_size; (void)ws_size;
}